// NodeCoordDecoder_26594437496849
// MI455X (gfx1250) — compile-verified
//
#include <hip/hip_runtime.h>

typedef __attribute__((ext_vector_type(2))) float v2f;
typedef __attribute__((ext_vector_type(8))) float v8f;

#define D 64
#define LN_EPS 1e-5f

// ---------------- degree: deg = 1 (self loop) + in-degree; then dinv = rsqrt(deg) ----
__global__ void k_deg_init(float* __restrict__ deg, int n) {
  int i = blockIdx.x * blockDim.x + threadIdx.x;
  if (i < n) deg[i] = 1.0f;
}
__global__ void k_deg_edges(const int* __restrict__ dst, float* __restrict__ deg, int e) {
  int i = blockIdx.x * blockDim.x + threadIdx.x;
  if (i < e) atomicAdd(&deg[dst[i]], 1.0f);
}
__global__ void k_deg_rsqrt(float* __restrict__ deg, int n) {
  int i = blockIdx.x * blockDim.x + threadIdx.x;
  if (i < n) deg[i] = rsqrtf(deg[i]);
}

// ---------------- fp32 WMMA GEMM: Y[N x 64] = X[N x 64] * W[64 x 64] (+bias, relu) ----
// One wave32 per 16-row tile; 4 column tiles of 16; K stepped by 4 via
// V_WMMA_F32_16X16X4_F32.  A layout (16x4 f32): lanes 0-15 row M=lane hold
// {K=k0,K=k0+1}, lanes 16-31 hold {K=k0+2,K=k0+3}.  B (4x16) mirrors with
// N=lane&15.  C/D: VGPR r -> row r (lanes 0-15) / row 8+r (lanes 16-31).
__global__ __launch_bounds__(128) void k_gemm64(const float* __restrict__ X,
                                                const float* __restrict__ W,
                                                const float* __restrict__ bias,
                                                float* __restrict__ Y,
                                                int n, int relu) {
  int ntiles = (n + 15) >> 4;
  int tile = blockIdx.x * 4 + (threadIdx.x >> 5);   // wave-uniform
  if (tile >= ntiles) return;                       // whole wave exits: EXEC stays full
  int lane = threadIdx.x & 31;
  int half = lane >> 4;      // 0 or 1 -> K sub-pair
  int mn   = lane & 15;      // row index for A, col index for B

  v8f acc0 = {}, acc1 = {}, acc2 = {}, acc3 = {};

  int row  = tile * 16 + mn;
  int rowc = row < n ? row : (n - 1);               // clamp tail loads, keep EXEC full
  const float* xrow = X + (size_t)rowc * D;

#pragma unroll
  for (int k0 = 0; k0 < D; k0 += 4) {
    int kb = k0 + 2 * half;
    v2f a;
    a.x = xrow[kb];
    a.y = xrow[kb + 1];
    const float* w0 = W + (size_t)kb * D + mn;      // row kb
    const float* w1 = w0 + D;                       // row kb+1
    v2f b;
    b.x = w0[0];  b.y = w1[0];
    acc0 = __builtin_amdgcn_wmma_f32_16x16x4_f32(false, a, false, b, (short)0, acc0, false, false);
    b.x = w0[16]; b.y = w1[16];
    acc1 = __builtin_amdgcn_wmma_f32_16x16x4_f32(false, a, false, b, (short)0, acc1, false, false);
    b.x = w0[32]; b.y = w1[32];
    acc2 = __builtin_amdgcn_wmma_f32_16x16x4_f32(false, a, false, b, (short)0, acc2, false, false);
    b.x = w0[48]; b.y = w1[48];
    acc3 = __builtin_amdgcn_wmma_f32_16x16x4_f32(false, a, false, b, (short)0, acc3, false, false);
  }

  float bb0 = 0.f, bb1 = 0.f, bb2 = 0.f, bb3 = 0.f;
  if (bias) { bb0 = bias[mn]; bb1 = bias[16 + mn]; bb2 = bias[32 + mn]; bb3 = bias[48 + mn]; }

#pragma unroll
  for (int r = 0; r < 8; ++r) {
    int orow = tile * 16 + half * 8 + r;
    if (orow < n) {
      float v0 = acc0[r] + bb0, v1 = acc1[r] + bb1, v2 = acc2[r] + bb2, v3 = acc3[r] + bb3;
      if (relu) {
        v0 = fmaxf(v0, 0.f); v1 = fmaxf(v1, 0.f);
        v2 = fmaxf(v2, 0.f); v3 = fmaxf(v3, 0.f);
      }
      float* yp = Y + (size_t)orow * D + mn;
      yp[0] = v0; yp[16] = v1; yp[32] = v2; yp[48] = v3;
    }
  }
}

// ---------------- self-loop init: Out = dinv^2 * H + b ----------------
__global__ void k_agg_init(const float* __restrict__ H, const float* __restrict__ dinv,
                           const float* __restrict__ b, float* __restrict__ Out, int n) {
  int t = blockIdx.x * blockDim.x + threadIdx.x;
  if (t >= n * D) return;
  int i = t >> 6, f = t & 63;
  float di = dinv[i];
  Out[t] = di * di * H[t] + b[f];
}

// ---------------- edge scatter: Out[dst] += dinv[src]*dinv[dst] * H[src] ----------
// One wave32 per edge; each lane handles a float2 (coalesced 256B row per edge).
// 25.6MB destination lives in the 192MB L2, so atomic_add_f32 stays on-chip.
__global__ __launch_bounds__(256) void k_scatter(const int* __restrict__ src,
                                                 const int* __restrict__ dst,
                                                 const float* __restrict__ dinv,
                                                 const float* __restrict__ H,
                                                 float* __restrict__ Out, int e) {
  int edge = blockIdx.x * 8 + (threadIdx.x >> 5);
  if (edge >= e) return;
  int lane = threadIdx.x & 31;
  int s = src[edge], d = dst[edge];
  float c = dinv[s] * dinv[d];
  float2 hv = *((const float2*)(H + (size_t)s * D) + lane);
  float* op = Out + (size_t)d * D + lane * 2;
  atomicAdd(op,     hv.x * c);
  atomicAdd(op + 1, hv.y * c);
}

// ---------------- ReLU + LayerNorm over 64 features, one wave per node ----------
__global__ __launch_bounds__(256) void k_relu_ln(const float* __restrict__ In,
                                                 const float* __restrict__ g,
                                                 const float* __restrict__ beta,
                                                 float* __restrict__ Out, int n) {
  int node = blockIdx.x * 8 + (threadIdx.x >> 5);
  if (node >= n) return;
  int lane = threadIdx.x & 31;
  float2 v = *((const float2*)(In + (size_t)node * D) + lane);
  v.x = fmaxf(v.x, 0.f); v.y = fmaxf(v.y, 0.f);
  float s  = v.x + v.y;
  float s2 = v.x * v.x + v.y * v.y;
#pragma unroll
  for (int off = 16; off >= 1; off >>= 1) {
    s  += __shfl_xor(s,  off, 32);
    s2 += __shfl_xor(s2, off, 32);
  }
  float mu   = s * (1.0f / 64.0f);
  float var  = s2 * (1.0f / 64.0f) - mu * mu;
  float rinv = rsqrtf(var + LN_EPS);
  float2 o;
  o.x = (v.x - mu) * rinv * g[lane * 2]     + beta[lane * 2];
  o.y = (v.y - mu) * rinv * g[lane * 2 + 1] + beta[lane * 2 + 1];
  *((float2*)(Out + (size_t)node * D) + lane) = o;
}

// ---------------- final projection: out[N x 3] = H[N x 64] @ Wm2 + bm2 ----------
__global__ void k_mlp_out(const float* __restrict__ H, const float* __restrict__ W,
                          const float* __restrict__ b, float* __restrict__ Out, int n) {
  int i = blockIdx.x * blockDim.x + threadIdx.x;
  if (i >= n) return;
  float a0 = b[0], a1 = b[1], a2 = b[2];
  const float* h = H + (size_t)i * D;
#pragma unroll
  for (int k = 0; k < D; ++k) {
    float hv = h[k];
    a0 += hv * W[k * 3 + 0];
    a1 += hv * W[k * 3 + 1];
    a2 += hv * W[k * 3 + 2];
  }
  Out[i * 3 + 0] = a0;
  Out[i * 3 + 1] = a1;
  Out[i * 3 + 2] = a2;
}

extern "C" void kernel_launch(void* const* d_in, const int* in_sizes, int n_in,
                              void* d_out, int out_size, void* d_ws, size_t ws_size,
                              hipStream_t stream) {
  const float* z   = (const float*)d_in[0];
  const int*   ei  = (const int*)  d_in[1];
  const float* W1  = (const float*)d_in[2];
  const float* b1  = (const float*)d_in[3];
  const float* g1  = (const float*)d_in[4];
  const float* be1 = (const float*)d_in[5];
  const float* W2  = (const float*)d_in[6];
  const float* b2  = (const float*)d_in[7];
  const float* g2  = (const float*)d_in[8];
  const float* be2 = (const float*)d_in[9];
  const float* Wm1 = (const float*)d_in[10];
  const float* bm1 = (const float*)d_in[11];
  const float* Wm2 = (const float*)d_in[12];
  const float* bm2 = (const float*)d_in[13];
  float* out = (float*)d_out;

  int n = in_sizes[0] / D;      // 100000
  int e = in_sizes[1] / 2;      // 1600000
  const int* src = ei;
  const int* dst = ei + e;

  // workspace: dinv [n] | A [n*64] | B [n*64]
  char* ws = (char*)d_ws;
  size_t off0 = ((size_t)n * sizeof(float) + 255) & ~(size_t)255;
  size_t szNF = ((size_t)n * D * sizeof(float) + 255) & ~(size_t)255;
  float* dinv = (float*)ws;
  float* A = (float*)(ws + off0);
  float* B = (float*)(ws + off0 + szNF);

  int tN  = (n + 255) / 256;
  int tE  = (e + 255) / 256;
  int tNF = (int)(((size_t)n * D + 255) / 256);
  int ntiles = (n + 15) / 16;
  int gemm_blocks = (ntiles + 3) / 4;
  int wave_n = (n + 7) / 8;
  int wave_e = (e + 7) / 8;

  // symmetric-norm degrees
  k_deg_init <<<tN, 256, 0, stream>>>(dinv, n);
  k_deg_edges<<<tE, 256, 0, stream>>>(dst, dinv, e);
  k_deg_rsqrt<<<tN, 256, 0, stream>>>(dinv, n);

  // GCN layer 1: A = z@W1 ; B = dinv^2*A + b1 ; B += scatter ; A = LN(relu(B))
  k_gemm64  <<<gemm_blocks, 128, 0, stream>>>(z, W1, nullptr, A, n, 0);
  k_agg_init<<<tNF, 256, 0, stream>>>(A, dinv, b1, B, n);
  k_scatter <<<wave_e, 256, 0, stream>>>(src, dst, dinv, A, B, e);
  k_relu_ln <<<wave_n, 256, 0, stream>>>(B, g1, be1, A, n);

  // GCN layer 2: B = A@W2 ; A = dinv^2*B + b2 ; A += scatter ; B = LN(relu(A))
  k_gemm64  <<<gemm_blocks, 128, 0, stream>>>(A, W2, nullptr, B, n, 0);
  k_agg_init<<<tNF, 256, 0, stream>>>(B, dinv, b2, A, n);
  k_scatter <<<wave_e, 256, 0, stream>>>(src, dst, dinv, B, A, e);
  k_relu_ln <<<wave_n, 256, 0, stream>>>(A, g2, be2, B, n);

  // MLP: A = relu(B@Wm1 + bm1) ; out = A@Wm2 + bm2
  k_gemm64 <<<gemm_blocks, 128, 0, stream>>>(B, Wm1, bm1, A, n, 1);
  k_mlp_out<<<tN, 256, 0, stream>>>(A, Wm2, bm2, out, n);
}